// BregmanADMM12_3624952398645
// MI455X (gfx1250) — compile-verified
//
#include <hip/hip_runtime.h>
#include <stdint.h>

#define TOT_ELEMS (16u * 1024u * 1024u)   // 16 x 1024 x 1024
#define NROWS     16384                   // B*M rows
#define NCOL      1024                    // softmax axis
#define RPB       8                       // rows per block
#define NBLK      (NROWS / RPB)           // 2048 blocks
#define THREADS   256                     // 8 wave32 waves

typedef unsigned int v4u __attribute__((ext_vector_type(4)));
typedef int          v8i __attribute__((ext_vector_type(8)));
typedef int          v4i __attribute__((ext_vector_type(4)));
typedef float        v2f __attribute__((ext_vector_type(2)));
typedef float        v8f __attribute__((ext_vector_type(8)));

// ---------------- wave32 / block reductions ----------------
__device__ __forceinline__ float wave_max(float v) {
#pragma unroll
  for (int m = 16; m >= 1; m >>= 1) v = fmaxf(v, __shfl_xor(v, m, 32));
  return v;
}
__device__ __forceinline__ float wave_sum(float v) {
#pragma unroll
  for (int m = 16; m >= 1; m >>= 1) v += __shfl_xor(v, m, 32);
  return v;
}
__device__ __forceinline__ float block_max(float v, float* red) {
  v = wave_max(v);
  if ((threadIdx.x & 31) == 0) red[threadIdx.x >> 5] = v;
  __syncthreads();
  float r = red[0];
#pragma unroll
  for (int i = 1; i < THREADS / 32; ++i) r = fmaxf(r, red[i]);
  __syncthreads();
  return r;
}
__device__ __forceinline__ float block_sum(float v, float* red) {
  v = wave_sum(v);
  if ((threadIdx.x & 31) == 0) red[threadIdx.x >> 5] = v;
  __syncthreads();
  float r = red[0];
#pragma unroll
  for (int i = 1; i < THREADS / 32; ++i) r += red[i];
  __syncthreads();
  return r;
}

// ---------------- prologue: L0 = log(where(R0==0,1e-40,R0)), Z = 0 ----------------
__global__ __launch_bounds__(THREADS)
void bregman_prologue(const float* __restrict__ R0, float* __restrict__ L0,
                      float* __restrict__ Z) {
  const size_t stride = (size_t)gridDim.x * blockDim.x;
  const size_t n4 = TOT_ELEMS / 4;
  for (size_t i = (size_t)blockIdx.x * blockDim.x + threadIdx.x; i < n4; i += stride) {
    float4 r = ((const float4*)R0)[i];
    float4 o;
    o.x = __logf(r.x == 0.f ? 1e-40f : r.x);
    o.y = __logf(r.y == 0.f ? 1e-40f : r.y);
    o.z = __logf(r.z == 0.f ? 1e-40f : r.z);
    o.w = __logf(r.w == 0.f ? 1e-40f : r.w);
    ((float4*)L0)[i] = o;
    ((float4*)Z)[i]  = make_float4(0.f, 0.f, 0.f, 0.f);
  }
}

// ---------------- fused per-iteration kernel ----------------
// Entering iter t (t>=2): Rg=R_{t-1}, Zg=Z_{t-2}, scaleg=scale_{t-1}.
// Computes X_{t-1} on the fly, writes Z_{t-1}, then R_t = softmax(...) in place,
// and per-block column partial sums of s_td_t^2 (deterministic, no atomics).
__global__ __launch_bounds__(THREADS)
void bregman_iter(const float* __restrict__ L0g, float* __restrict__ Rg,
                  float* __restrict__ Zg, const float* __restrict__ scaleg,
                  float* __restrict__ partials, int first) {
  __shared__ float l0t[RPB][NCOL];          // 32 KB tile staged via TDM
  __shared__ float red[THREADS / 32];

  const int tid  = threadIdx.x;
  const int row0 = blockIdx.x * RPB;
  const int c0   = tid * 4;                 // 4 contiguous columns per thread

  // ---- stage L0 tile: Tensor Data Mover (async DMA -> LDS) ----
#if __has_builtin(__builtin_amdgcn_tensor_load_to_lds)
  if (tid < 32) {                           // wave 0 only (uniform branch)
    const unsigned long long ga =
        (unsigned long long)(uintptr_t)(L0g + (size_t)row0 * NCOL);
    const unsigned int ldsoff = (unsigned int)(uintptr_t)(&l0t[0][0]);
    v4u g0;
    g0[0] = 1u;                                            // count = 1 (valid D#)
    g0[1] = ldsoff;                                        // lds_addr
    g0[2] = (unsigned int)ga;                              // global_addr[31:0]
    g0[3] = (unsigned int)((ga >> 32) & 0x1FFFFFFull) | (2u << 30); // addr[56:32] | type=2
    v8i g1;
    g1[0] = (int)(2u << 16);                               // data_size = 4B
    g1[1] = (int)(((unsigned)NCOL & 0xFFFFu) << 16);       // tensor_dim0 lo16
    g1[2] = (int)((((unsigned)NCOL >> 16) & 0xFFFFu) |
                  (((unsigned)NROWS & 0xFFFFu) << 16));    // dim0 hi | dim1 lo
    g1[3] = (int)(((((unsigned)NROWS) >> 16) & 0xFFFFu) |
                  (((unsigned)NCOL & 0xFFFFu) << 16));     // dim1 hi | tile_dim0
    g1[4] = RPB;                                           // tile_dim1 (tile_dim2=0)
    g1[5] = NCOL;                                          // tensor_dim0_stride lo32
    g1[6] = 0;
    g1[7] = 0;
    v4i g2 = {0, 0, 0, 0};
    v4i g3 = {0, 0, 0, 0};
#if defined(__clang_major__) && (__clang_major__ >= 23)
    v8i g4 = {0, 0, 0, 0, 0, 0, 0, 0};
    __builtin_amdgcn_tensor_load_to_lds(g0, g1, g2, g3, g4, 0);
#else
    __builtin_amdgcn_tensor_load_to_lds(g0, g1, g2, g3, 0);
#endif
#if __has_builtin(__builtin_amdgcn_s_wait_tensorcnt)
    __builtin_amdgcn_s_wait_tensorcnt(0);
#endif
  }
  __syncthreads();
#else
  for (int idx = tid; idx < RPB * NCOL / 4; idx += THREADS)
    ((float4*)&l0t[0][0])[idx] = ((const float4*)(L0g + (size_t)row0 * NCOL))[idx];
  __syncthreads();
#endif

  float scv[4] = {0.f, 0.f, 0.f, 0.f};
  if (!first) {
    const float4 s4 = *(const float4*)(scaleg + c0);
    scv[0] = s4.x; scv[1] = s4.y; scv[2] = s4.z; scv[3] = s4.w;
  }

  float acc[4] = {0.f, 0.f, 0.f, 0.f};      // column sums of s_td^2

  for (int i = 0; i < RPB; ++i) {
    const size_t base = (size_t)(row0 + i) * NCOL + c0;
    const float4 l04 = *(const float4*)&l0t[i][c0];
    const float l0[4] = {l04.x, l04.y, l04.z, l04.w};

    float tmp[4], zn[4];
    if (first) {
      // carry init: X=R_0, Z=0 -> tmp = (L0 + L0)/2 = L0, Z stays 0
#pragma unroll
      for (int c = 0; c < 4; ++c) { tmp[c] = l0[c]; zn[c] = 0.f; }
    } else {
      const float4 r4 = *(const float4*)(Rg + base);
      const float4 z4 = *(const float4*)(Zg + base);
      const float rv[4] = {r4.x, r4.y, r4.z, r4.w};
      const float zv[4] = {z4.x, z4.y, z4.z, z4.w};
#pragma unroll
      for (int c = 0; c < 4; ++c) {
        const float x  = rv[c] + zv[c];                  // R + Z/rho (rho=1)
        const float a  = fabsf(x) - 0.05f;               // alpha*lambda/rho
        const float st = (a > 0.f) ? copysignf(a, x) : 0.f;
        const float X  = scv[c] * st;                    // group shrink
        const float Xs = (X == 0.f) ? 1e-40f : X;        // MIN_VALUE guard
        const float z2 = zv[c] + rv[c] - X;              // Z_new = Z + rho*(R - X)
        zn[c]  = z2;
        tmp[c] = 0.5f * (l0[c] + __logf(Xs) - z2);       // /(1+rho)
      }
      *(float4*)(Zg + base) = make_float4(zn[0], zn[1], zn[2], zn[3]);
    }

    // row softmax
    float m = fmaxf(fmaxf(tmp[0], tmp[1]), fmaxf(tmp[2], tmp[3]));
    m = block_max(m, red);
    float e[4], p = 0.f;
#pragma unroll
    for (int c = 0; c < 4; ++c) { e[c] = __expf(tmp[c] - m); p += e[c]; }
    const float S   = block_sum(p, red);
    const float inv = 1.0f / S;
#pragma unroll
    for (int c = 0; c < 4; ++c) e[c] *= inv;
    *(float4*)(Rg + base) = make_float4(e[0], e[1], e[2], e[3]);

    // s_td of the NEW R (for this iteration's column norms)
#pragma unroll
    for (int c = 0; c < 4; ++c) {
      const float x2 = e[c] + zn[c];
      const float a2 = fabsf(x2) - 0.05f;
      const float s2 = (a2 > 0.f) ? a2 : 0.f;            // |s_td|
      acc[c] = fmaf(s2, s2, acc[c]);
    }
  }

  *(float4*)(partials + (size_t)blockIdx.x * NCOL + c0) =
      make_float4(acc[0], acc[1], acc[2], acc[3]);
}

// ---------------- per-column scale via WMMA ones-GEMV ----------------
// colsum[n] = sum_b partials[b][n] == (ones(16x4) x B-tiles) accumulated with
// V_WMMA_F32_16X16X4_F32. With A == all-ones, D[m][n] = sum_k B[k][n]: the sum
// is invariant to K-slot permutation in the B layout; only the documented
// "column n in lanes n and n+16" mapping matters. Full f32 precision,
// deterministic fixed-order accumulation. 64 waves cover the 1024 columns.
__global__ __launch_bounds__(THREADS)
void bregman_colscale(const float* __restrict__ partials, float* __restrict__ scaleg) {
  const int lane = threadIdx.x & 31;
  const int wid  = (blockIdx.x * THREADS + threadIdx.x) >> 5;  // 0..63
  const int col  = wid * 16 + (lane & 15);
  const int rowA = (lane >> 4) * 2;        // lanes 0-15 -> rows b0,b0+1; 16-31 -> b0+2,b0+3

  v2f a; a.x = 1.0f; a.y = 1.0f;           // all-ones A (16x4)
  v8f c = {0.f, 0.f, 0.f, 0.f, 0.f, 0.f, 0.f, 0.f};

  for (int b0 = 0; b0 < NBLK; b0 += 4) {
    v2f b;
    b.x = partials[(size_t)(b0 + rowA)     * NCOL + col];
    b.y = partials[(size_t)(b0 + rowA + 1) * NCOL + col];
    c = __builtin_amdgcn_wmma_f32_16x16x4_f32(false, a, false, b,
                                              (short)0, c, false, false);
  }

  if (lane < 16) {
    const float s    = c[0];               // row M=0 of D: colsum for `col`
    const float norm = sqrtf(s);
    // scale = max(1 - (1-alpha)*lambda/(rho*norm + eps), 0) = max(1 - 0.05/(norm+1e-10), 0)
    scaleg[col] = fmaxf(1.0f - 0.05f / (norm + 1e-10f), 0.0f);
  }
}

extern "C" void kernel_launch(void* const* d_in, const int* in_sizes, int n_in,
                              void* d_out, int out_size, void* d_ws, size_t ws_size,
                              hipStream_t stream) {
  (void)in_sizes; (void)n_in; (void)out_size; (void)ws_size;

  const float* R0 = (const float*)d_in[0];
  float* R = (float*)d_out;                 // R lives in d_out (final answer = R_20)

  char* ws = (char*)d_ws;
  float* L0       = (float*)(ws);                                   // 64 MB
  float* Z        = (float*)(ws + (size_t)TOT_ELEMS * 4);           // 64 MB
  float* partials = (float*)(ws + (size_t)TOT_ELEMS * 8);           // 8 MB
  float* scale    = (float*)(ws + (size_t)TOT_ELEMS * 8 +
                             (size_t)NBLK * NCOL * 4);              // 4 KB

  bregman_prologue<<<4096, THREADS, 0, stream>>>(R0, L0, Z);

  for (int it = 0; it < 20; ++it) {
    bregman_iter<<<NBLK, THREADS, 0, stream>>>(L0, R, Z, scale, partials, it == 0);
    if (it < 19)  // last iteration's scale is never consumed
      bregman_colscale<<<8, THREADS, 0, stream>>>(partials, scale);
  }
}